// GATEncoder_74028056314067
// MI455X (gfx1250) — compile-verified
//
#include <hip/hip_runtime.h>
#include <hip/hip_bf16.h>
#include <math.h>

#define NEG_SLOPE 0.2f

typedef float v2f __attribute__((ext_vector_type(2)));
typedef float v8f __attribute__((ext_vector_type(8)));

// gfx1250 async global->LDS copy path (guarded; falls back to plain LDS copy).
#ifdef __has_builtin
#if __has_builtin(__builtin_amdgcn_global_load_async_to_lds_b128) && \
    __has_builtin(__builtin_amdgcn_s_wait_asynccnt)
#define USE_ASYNC_LDS 1
#endif
#endif
#ifndef USE_ASYNC_LDS
#define USE_ASYNC_LDS 0
#endif

#if USE_ASYNC_LDS
// Builtin signature (from hipcc diagnostic): param0 = v4i in AS1 (global),
// param1 = v4i in AS3 (LDS), then imm offset, imm cpol.
typedef int v4i __attribute__((vector_size(16)));
typedef __attribute__((address_space(1))) v4i* v4i_gp;
typedef __attribute__((address_space(3))) v4i* v4i_lp;
#endif

#define MAXC 256
#define LDS_PAD 4            // 260-dword row stride -> conflict-free ds_load_b64
#define XS_STRIDE (MAXC + LDS_PAD)

// ---------------------------------------------------------------------------
// GEMM: H[n x Cout] = X[n x Cin] @ W[Cin x Cout]
// Block = 256 threads (8 waves). A block owns a 16-row strip of X, staged once
// into LDS (async global->LDS when available), then each wave computes one
// 16x16 tile of a 128-wide column group via V_WMMA_F32_16X16X4_F32.
// Fragment layouts per CDNA5 ISA 7.12.2:
//   A (16x4):  lanes 0-15 -> row M=lane, VGPR0=K0 VGPR1=K1;
//              lanes 16-31 -> row M=lane-16, VGPR0=K2 VGPR1=K3.
//   B (4x16):  VGPR0: K0 (lanes 0-15) / K2 (lanes 16-31); VGPR1: K1 / K3.
//   C (16x16): VGPR r -> M=r (lanes 0-15) / M=r+8 (lanes 16-31).
// Requires n%16==0, Cout%128==0, Cin%4==0 (holds: 50000 / {128,256} / {128,256}).
// ---------------------------------------------------------------------------
__global__ __launch_bounds__(256) void gat_gemm_wmma(const float* __restrict__ X,
                                                     const float* __restrict__ W,
                                                     float* __restrict__ H,
                                                     int Cin, int Cout) {
  __shared__ float xs[16 * XS_STRIDE];

  const int tid  = threadIdx.x;
  const int lane = tid & 31;
  const int wave = tid >> 5;        // 0..7
  const int l15  = lane & 15;
  const int hi   = lane >> 4;       // 0 -> K pair {0,1}, 1 -> K pair {2,3}
  const int m0   = blockIdx.x * 16;
  const int n0   = blockIdx.y * 128 + wave * 16;

  // ---- Stage X[m0:m0+16, 0:Cin] into LDS (16B chunks, coalesced) ----
  const int cpr = Cin >> 2;                 // float4 chunks per row
  for (int c = tid; c < 16 * cpr; c += 256) {
    const int row = c / cpr, col4 = c - row * cpr;
    const float* g = X + (size_t)(m0 + row) * Cin + col4 * 4;
    float*       l = &xs[row * XS_STRIDE + col4 * 4];
#if USE_ASYNC_LDS
    // AS1 ptr = 64-bit flat VA; AS3 ptr = low 32 bits of flat VA (LDS offset).
    v4i_gp gp = (v4i_gp)(unsigned long long)(size_t)g;
    v4i_lp lp = (v4i_lp)(unsigned int)(size_t)l;
    __builtin_amdgcn_global_load_async_to_lds_b128(gp, lp, 0, 0);
#else
    *(float4*)l = *(const float4*)g;
#endif
  }
#if USE_ASYNC_LDS
  __builtin_amdgcn_s_wait_asynccnt(0);
#endif
  __syncthreads();

  // ---- WMMA K-loop: A from LDS (bank-conflict-free), B from global (L2) ----
  const float* xrow = &xs[l15 * XS_STRIDE + 2 * hi];
  const float* wcol = W + (size_t)(2 * hi) * Cout + n0 + l15;

  v8f acc = {};
  for (int k = 0; k < Cin; k += 4) {
    v2f a, b;
    a.x = xrow[k];
    a.y = xrow[k + 1];
    b.x = wcol[(size_t)k * Cout];
    b.y = wcol[(size_t)(k + 1) * Cout];
    acc = __builtin_amdgcn_wmma_f32_16x16x4_f32(false, a, false, b,
                                                (short)0, acc, false, false);
  }

  float* orow = H + (size_t)(m0 + 8 * hi) * Cout + n0 + l15;
#pragma unroll
  for (int r = 0; r < 8; ++r) orow[(size_t)r * Cout] = acc[r];
}

// ---------------------------------------------------------------------------
// Order-preserving float<->uint key for atomicMax-based segment max.
// memset(0) initializes keys to the minimum element.
// ---------------------------------------------------------------------------
__device__ __forceinline__ unsigned f2key(float f) {
  int i = __float_as_int(f);
  unsigned u = (unsigned)i;
  return (i >= 0) ? (u | 0x80000000u) : ~u;
}
__device__ __forceinline__ float key2f(unsigned u) {
  int i = (u & 0x80000000u) ? (int)(u & 0x7FFFFFFFu) : (int)~u;
  return __int_as_float(i);
}

// Per-node attention logits: s_src[i] = h[i,:].a_src, s_dst[i] = h[i,:].a_dst.
__global__ __launch_bounds__(256) void gat_att_scores(const float* __restrict__ H,
                                                      const float* __restrict__ a_src,
                                                      const float* __restrict__ a_dst,
                                                      float* __restrict__ s_src,
                                                      float* __restrict__ s_dst,
                                                      int n, int C) {
  const int lane = threadIdx.x & 31;
  const int node = blockIdx.x * 8 + (threadIdx.x >> 5);
  if (node >= n) return;
  const float* h = H + (size_t)node * C;
  float as = 0.f, ad = 0.f;
  for (int c = lane; c < C; c += 32) {
    float v = h[c];
    as += v * a_src[c];
    ad += v * a_dst[c];
  }
#pragma unroll
  for (int off = 16; off; off >>= 1) {
    as += __shfl_xor(as, off, 32);
    ad += __shfl_xor(ad, off, 32);
  }
  if (lane == 0) {
    s_src[node] = as;
    s_dst[node] = ad;
  }
}

// Edge logits + running segment max over dst (atomicMax on uint keys in L2).
__global__ __launch_bounds__(256) void gat_edge_logits_max(const long long* __restrict__ srcv,
                                                           const long long* __restrict__ dstv,
                                                           const float* __restrict__ s_src,
                                                           const float* __restrict__ s_dst,
                                                           float* __restrict__ ebuf,
                                                           unsigned* __restrict__ mkey,
                                                           int E) {
  int e = blockIdx.x * blockDim.x + threadIdx.x;
  if (e >= E) return;
  int s = (int)srcv[e];
  int d = (int)dstv[e];
  float v = s_src[s] + s_dst[d];
  v = (v > 0.f) ? v : NEG_SLOPE * v;  // leaky_relu
  ebuf[e] = v;
  atomicMax(mkey + d, f2key(v));
}

// p = exp(e - max[dst]); denom[dst] += p.
__global__ __launch_bounds__(256) void gat_edge_exp_sum(const long long* __restrict__ dstv,
                                                        const unsigned* __restrict__ mkey,
                                                        float* __restrict__ ebuf,
                                                        float* __restrict__ denom,
                                                        int E) {
  int e = blockIdx.x * blockDim.x + threadIdx.x;
  if (e >= E) return;
  int d = (int)dstv[e];
  float p = __expf(ebuf[e] - key2f(mkey[d]));
  ebuf[e] = p;
  atomicAdd(denom + d, p);
}

// out[dst,:] += (p/denom[dst]) * h[src,:]   (one wave per edge; h is L2-resident)
__global__ __launch_bounds__(256) void gat_edge_aggregate(const long long* __restrict__ srcv,
                                                          const long long* __restrict__ dstv,
                                                          const float* __restrict__ pbuf,
                                                          const float* __restrict__ denom,
                                                          const float* __restrict__ H,
                                                          float* __restrict__ out,
                                                          int E, int C) {
  const int lane = threadIdx.x & 31;
  const int e = blockIdx.x * 8 + (threadIdx.x >> 5);
  if (e >= E) return;
  int s = (int)srcv[e];
  int d = (int)dstv[e];
  float alpha = pbuf[e] / denom[d];
  const float* hs = H + (size_t)s * C;
  float* od = out + (size_t)d * C;
  for (int c = lane; c < C; c += 32) atomicAdd(od + c, alpha * hs[c]);
}

__global__ __launch_bounds__(256) void gat_bias_act(float* __restrict__ out,
                                                    const float* __restrict__ bias,
                                                    int n, int C, int relu) {
  size_t i = (size_t)blockIdx.x * blockDim.x + threadIdx.x;
  if (i >= (size_t)n * C) return;
  int c = (int)(i % C);
  float v = out[i] + bias[c];
  if (relu) v = fmaxf(v, 0.f);
  out[i] = v;
}

// Row softmax (C=128), one wave32 per row.
__global__ __launch_bounds__(256) void gat_row_softmax(const float* __restrict__ in,
                                                       float* __restrict__ out,
                                                       int n, int C) {
  const int lane = threadIdx.x & 31;
  const int row = blockIdx.x * 8 + (threadIdx.x >> 5);
  if (row >= n) return;
  const float* r = in + (size_t)row * C;
  float mx = -INFINITY;
  for (int c = lane; c < C; c += 32) mx = fmaxf(mx, r[c]);
#pragma unroll
  for (int off = 16; off; off >>= 1) mx = fmaxf(mx, __shfl_xor(mx, off, 32));
  float sum = 0.f;
  for (int c = lane; c < C; c += 32) sum += __expf(r[c] - mx);
#pragma unroll
  for (int off = 16; off; off >>= 1) sum += __shfl_xor(sum, off, 32);
  float inv = 1.f / sum;
  for (int c = lane; c < C; c += 32) out[(size_t)row * C + c] = __expf(r[c] - mx) * inv;
}

// ---------------------------------------------------------------------------
extern "C" void kernel_launch(void* const* d_in, const int* in_sizes, int n_in,
                              void* d_out, int out_size, void* d_ws, size_t ws_size,
                              hipStream_t stream) {
  (void)n_in; (void)out_size; (void)ws_size;

  const float*     x   = (const float*)d_in[0];
  const long long* ei  = (const long long*)d_in[1];  // int64 edge_index [2, E]
  const float* W1  = (const float*)d_in[2];
  const float* a1s = (const float*)d_in[3];
  const float* a1d = (const float*)d_in[4];
  const float* b1  = (const float*)d_in[5];
  const float* W2  = (const float*)d_in[6];
  const float* a2s = (const float*)d_in[7];
  const float* a2d = (const float*)d_in[8];
  const float* b2  = (const float*)d_in[9];
  const float* W3  = (const float*)d_in[10];
  const float* a3s = (const float*)d_in[11];
  const float* a3d = (const float*)d_in[12];
  const float* b3  = (const float*)d_in[13];

  const int DIN = 128, HCH = 256, DOUT = 128;
  const int n = in_sizes[0] / DIN;   // 50000
  const int E = in_sizes[1] / 2;     // 1600000
  const long long* srcv = ei;
  const long long* dstv = ei + E;

  // Workspace bump allocator (256B aligned). Total ~110 MB.
  char* p = (char*)d_ws;
  auto alloc = [&](size_t bytes) -> char* {
    char* r = p;
    p += (bytes + 255) & ~(size_t)255;
    return r;
  };
  float*    hA    = (float*)alloc((size_t)n * HCH * sizeof(float));
  float*    hB    = (float*)alloc((size_t)n * HCH * sizeof(float));
  float*    ssrc  = (float*)alloc((size_t)n * sizeof(float));
  float*    sdst  = (float*)alloc((size_t)n * sizeof(float));
  unsigned* mkey  = (unsigned*)alloc((size_t)n * sizeof(unsigned));
  float*    denom = (float*)alloc((size_t)n * sizeof(float));
  float*    ebuf  = (float*)alloc((size_t)E * sizeof(float));

  auto conv = [&](const float* in, int Cin, int Cout, const float* W,
                  const float* as, const float* ad, const float* bias,
                  float* h, float* out, bool relu) {
    dim3 g(n / 16, Cout / 128);
    gat_gemm_wmma<<<g, 256, 0, stream>>>(in, W, h, Cin, Cout);
    (void)hipMemsetAsync(out, 0, (size_t)n * Cout * sizeof(float), stream);
    (void)hipMemsetAsync(mkey, 0, (size_t)n * sizeof(unsigned), stream);
    (void)hipMemsetAsync(denom, 0, (size_t)n * sizeof(float), stream);
    gat_att_scores<<<(n + 7) / 8, 256, 0, stream>>>(h, as, ad, ssrc, sdst, n, Cout);
    gat_edge_logits_max<<<(E + 255) / 256, 256, 0, stream>>>(srcv, dstv, ssrc, sdst,
                                                             ebuf, mkey, E);
    gat_edge_exp_sum<<<(E + 255) / 256, 256, 0, stream>>>(dstv, mkey, ebuf, denom, E);
    gat_edge_aggregate<<<(E + 7) / 8, 256, 0, stream>>>(srcv, dstv, ebuf, denom,
                                                        h, out, E, Cout);
    gat_bias_act<<<(unsigned)(((size_t)n * Cout + 255) / 256), 256, 0, stream>>>(
        out, bias, n, Cout, relu ? 1 : 0);
  };

  // Layer 1: 128 -> 256, relu
  conv(x,  DIN, HCH, W1, a1s, a1d, b1, hA, hB, true);
  // Layers 2 & 3 share conv2 weights: 256 -> 256, relu
  conv(hB, HCH, HCH, W2, a2s, a2d, b2, hA, hB, true);
  conv(hB, HCH, HCH, W2, a2s, a2d, b2, hA, hB, true);
  // Layer 4: 256 -> 128, no relu
  conv(hB, HCH, DOUT, W3, a3s, a3d, b3, hA, hB, false);
  // Final row softmax -> d_out
  gat_row_softmax<<<(n + 7) / 8, 256, 0, stream>>>(hB, (float*)d_out, n, DOUT);
}